// TemperatureModel_RNN_72052371358097
// MI455X (gfx1250) — compile-verified
//
#include <hip/hip_runtime.h>
#include <hip/hip_bf16.h>
#include <math.h>

typedef __attribute__((ext_vector_type(2))) float v2f;
typedef __attribute__((ext_vector_type(8))) float v8f;

// OUT[M,N] = act( X[M,K] @ W[N,K]^T + bias1 + bias2 + addmat )
// Row-major. Requires K % 4 == 0, M % 16 == 0; N arbitrary (edge-guarded on store;
// edge lanes clamp their B row to N-1 and compute discarded garbage, keeping the
// inner loop branch-free / EXEC-stable).
// One wave32 computes one 16x16 tile via V_WMMA_F32_16X16X4_F32, two independent
// accumulators over even/odd K-chunks to break the WMMA C-chain.
// X == nullptr means X is the zero matrix (used for h_{-1} = 0).
__global__ __launch_bounds__(256) void gemm_xwt_wmma(
    const float* __restrict__ X, const float* __restrict__ W,
    const float* __restrict__ bias1, const float* __restrict__ bias2,
    const float* __restrict__ addmat, float* __restrict__ OUT,
    int M, int N, int K, int apply_tanh)
{
  const int lane   = threadIdx.x & 31;
  const int tilesN = (N + 15) >> 4;
  const int tilesM = M >> 4;
  const int tile   = blockIdx.x * (blockDim.x >> 5) + (threadIdx.x >> 5);
  if (tile >= tilesM * tilesN) return;
  const int m0 = (tile / tilesN) << 4;
  const int n0 = (tile % tilesN) << 4;

  const int  l15   = lane & 15;
  const int  khalf = (lane >> 4) << 1;   // lanes 0-15 hold K=0,1 ; lanes 16-31 hold K=2,3
  const int  j     = n0 + l15;           // output column owned by this lane (B & C/D layout)
  const bool jok   = (j < N);
  const int  jc    = jok ? j : (N - 1);  // clamped row: edge lanes compute discarded data

  float bval = 0.0f;
  if (jok && bias1) bval += bias1[j];
  if (jok && bias2) bval += bias2[j];
  v8f c, c1;
#pragma unroll
  for (int v = 0; v < 8; ++v) { c[v] = bval; c1[v] = 0.0f; }

  if (X) {
    const float* arow = X + (size_t)(m0 + l15) * K + khalf;   // A: 16x4 f32 fragment
    const float* brow = W + (size_t)jc * K + khalf;           // B: 4x16 f32 fragment
    int k0 = 0;
    for (; k0 + 8 <= K; k0 += 8) {
      v2f a0 = *(const v2f*)(arow + k0);
      v2f b0 = *(const v2f*)(brow + k0);
      v2f a1 = *(const v2f*)(arow + k0 + 4);
      v2f b1 = *(const v2f*)(brow + k0 + 4);
      c  = __builtin_amdgcn_wmma_f32_16x16x4_f32(false, a0, false, b0,
                                                 (short)0, c,  false, false);
      c1 = __builtin_amdgcn_wmma_f32_16x16x4_f32(false, a1, false, b1,
                                                 (short)0, c1, false, false);
    }
    if (k0 < K) {  // K % 8 == 4 tail
      v2f a0 = *(const v2f*)(arow + k0);
      v2f b0 = *(const v2f*)(brow + k0);
      c = __builtin_amdgcn_wmma_f32_16x16x4_f32(false, a0, false, b0,
                                                (short)0, c, false, false);
    }
#pragma unroll
    for (int v = 0; v < 8; ++v) c[v] += c1[v];
  }

  const int rhalf = (lane >> 4) << 3;    // lanes 0-15 -> rows m0+0..7 ; lanes 16-31 -> m0+8..15
  if (jok) {
    if (addmat) {
#pragma unroll
      for (int v = 0; v < 8; ++v)
        c[v] += addmat[(size_t)(m0 + rhalf + v) * N + j];
    }
    if (apply_tanh) {
#pragma unroll
      for (int v = 0; v < 8; ++v) c[v] = tanhf(c[v]);
    }
#pragma unroll
    for (int v = 0; v < 8; ++v)
      OUT[(size_t)(m0 + rhalf + v) * N + j] = c[v];
  }
}

// Layer 3 (H=20) full recurrence + final linear in one persistent workgroup.
// xp3: [S, 64, 20] with b_ih3+b_hh3 already added. out: [S, 64].
__global__ __launch_bounds__(640) void rnn3_linear_kernel(
    const float* __restrict__ xp3, const float* __restrict__ W_hh3,
    const float* __restrict__ W_lin, const float* __restrict__ b_lin,
    float* __restrict__ out, int S)
{
  const int B3 = 64, H3 = 20;
  __shared__ float h[2][64 * 20];
  __shared__ float w[20 * 20];
  __shared__ float wl[20];
  const int tid = threadIdx.x;

  if (tid < H3 * H3) w[tid] = W_hh3[tid];
  if (tid < H3)      wl[tid] = W_lin[tid];
  for (int i = tid; i < B3 * H3; i += blockDim.x) h[0][i] = 0.0f;
  __syncthreads();

  int cur = 0;
  for (int t = 0; t < S; ++t) {
    const float* xt = xp3 + (size_t)t * B3 * H3;
    for (int i = tid; i < B3 * H3; i += blockDim.x) {
      const int b = i / H3, jj = i % H3;
      float acc = xt[i];
#pragma unroll
      for (int k = 0; k < 20; ++k) acc += h[cur][b * H3 + k] * w[jj * H3 + k];
      h[cur ^ 1][i] = tanhf(acc);
    }
    __syncthreads();
    cur ^= 1;
    if (tid < B3) {
      float acc = b_lin[0];
#pragma unroll
      for (int k = 0; k < 20; ++k) acc += h[cur][tid * H3 + k] * wl[k];
      out[(size_t)t * B3 + tid] = acc;
    }
    __syncthreads();
  }
}

extern "C" void kernel_launch(void* const* d_in, const int* in_sizes, int n_in,
                              void* d_out, int out_size, void* d_ws, size_t ws_size,
                              hipStream_t stream) {
  const float* x     = (const float*)d_in[0];
  const float* W_ih1 = (const float*)d_in[1];
  const float* W_hh1 = (const float*)d_in[2];
  const float* b_ih1 = (const float*)d_in[3];
  const float* b_hh1 = (const float*)d_in[4];
  const float* W_ih2 = (const float*)d_in[5];
  const float* W_hh2 = (const float*)d_in[6];
  const float* b_ih2 = (const float*)d_in[7];
  const float* b_hh2 = (const float*)d_in[8];
  const float* W_ih3 = (const float*)d_in[9];
  const float* W_hh3 = (const float*)d_in[10];
  const float* b_ih3 = (const float*)d_in[11];
  const float* b_hh3 = (const float*)d_in[12];
  const float* W_lin = (const float*)d_in[13];
  const float* b_lin = (const float*)d_in[14];
  (void)in_sizes; (void)n_in; (void)out_size; (void)ws_size;

  const int S = 256, B = 64, I = 60, H1 = 1500, H2 = 3000, H3 = 20;
  const int MB = S * B; // 16384

  // Workspace: xp/hs buffers (recurrence overwrites xp[t] with h[t] in place).
  float* xp1 = (float*)d_ws;                 // [S*B, H1]  (98.3 MB)
  float* xp2 = xp1 + (size_t)MB * H1;        // [S*B, H2]  (196.6 MB)
  float* xp3 = xp2 + (size_t)MB * H2;        // [S*B, H3]  (1.3 MB)
  float* out = (float*)d_out;                // [S, B, 1]

  auto launch_gemm = [&](const float* X, const float* W, const float* b1,
                         const float* b2, const float* add, float* OUT,
                         int M, int N, int K, int tnh) {
    int tiles  = (M >> 4) * ((N + 15) >> 4);
    int blocks = (tiles + 7) / 8;            // 8 waves (256 threads) per block
    gemm_xwt_wmma<<<blocks, 256, 0, stream>>>(X, W, b1, b2, add, OUT, M, N, K, tnh);
  };

  // ---- Layer 1 ----
  launch_gemm(x, W_ih1, b_ih1, b_hh1, nullptr, xp1, MB, H1, I, 0);
  for (int t = 0; t < S; ++t) {
    const float* hprev = (t == 0) ? nullptr : (xp1 + (size_t)(t - 1) * B * H1);
    float* slot = xp1 + (size_t)t * B * H1;
    launch_gemm(hprev, W_hh1, nullptr, nullptr, slot, slot, B, H1, H1, 1);
  }
  // ---- Layer 2 ----
  launch_gemm(xp1, W_ih2, b_ih2, b_hh2, nullptr, xp2, MB, H2, H1, 0);
  for (int t = 0; t < S; ++t) {
    const float* hprev = (t == 0) ? nullptr : (xp2 + (size_t)(t - 1) * B * H2);
    float* slot = xp2 + (size_t)t * B * H2;
    launch_gemm(hprev, W_hh2, nullptr, nullptr, slot, slot, B, H2, H2, 1);
  }
  // ---- Layer 3 projection, then fused recurrence + linear ----
  launch_gemm(xp2, W_ih3, b_ih3, b_hh3, nullptr, xp3, MB, H3, H2, 0);
  rnn3_linear_kernel<<<1, 640, 0, stream>>>(xp3, W_hh3, W_lin, b_lin, out, S);
}